// SAGE_876173328847
// MI455X (gfx1250) — compile-verified
//
#include <hip/hip_runtime.h>
#include <hip/hip_bf16.h>

// ---------------------------------------------------------------------------
// 2-layer bipartite GraphSAGE (mean agg) for MI455X / gfx1250.
//  - Edge scatter-mean via global f32 atomics (tables are L2-resident: 192MB L2)
//  - Dense GEMMs via V_WMMA_F32_16X16X4_F32 (fp32, matches reference precision)
//  - Row-wise log_softmax with wave32 shfl reductions
// ---------------------------------------------------------------------------

#define N0_   200000
#define N1_   100000
#define N2_   20000
#define E1_   1600000
#define E2_   320000
#define D_IN_ 128
#define D_H_  256
#define D_OUT_ 64

typedef __attribute__((ext_vector_type(2))) float v2f;
typedef __attribute__((ext_vector_type(8))) float v8f;

// ---------------------------------------------------------------------------
// Zero-fill (accumulators must be re-zeroed on every launch / graph replay)
// ---------------------------------------------------------------------------
__global__ void zero_f4_kernel(float4* __restrict__ p, long n4) {
    long i = (long)blockIdx.x * blockDim.x + threadIdx.x;
    long stride = (long)gridDim.x * blockDim.x;
    float4 z = make_float4(0.f, 0.f, 0.f, 0.f);
    for (; i < n4; i += stride) p[i] = z;
}

// ---------------------------------------------------------------------------
// Edge scatter-sum + count.  One wave per edge; D/128 float4 chunks per lane.
// ---------------------------------------------------------------------------
template <int D>
__global__ void scatter_sum_kernel(const float* __restrict__ feat,
                                   const int* __restrict__ src,
                                   const int* __restrict__ dst,
                                   float* __restrict__ aggSum,
                                   float* __restrict__ cnt, int E) {
    int e = blockIdx.x * (blockDim.x >> 5) + (threadIdx.x >> 5);
    int lane = threadIdx.x & 31;
    if (e >= E) return;
    int s = src[e];
    int d = dst[e];
    const float4* row = reinterpret_cast<const float4*>(feat + (size_t)s * D);
    float* o = aggSum + (size_t)d * D;
#pragma unroll
    for (int c = 0; c < D / 128; ++c) {
        int idx = lane + 32 * c;              // float4 index within row
        float4 v = row[idx];
        float* op = o + idx * 4;
        atomicAdd(op + 0, v.x);
        atomicAdd(op + 1, v.y);
        atomicAdd(op + 2, v.z);
        atomicAdd(op + 3, v.w);
    }
    if (lane == 0) atomicAdd(cnt + d, 1.0f);
}

// ---------------------------------------------------------------------------
// Fused SAGE layer GEMM via WMMA f32 16x16x4:
//   out[m, n] = act( (aggSum[m,:]/max(cnt[m],1)) . Wl[n,:]
//                    + root[m,:] . Wr[n,:] + bias[n] )
// One wave owns one 16x16 output tile. EXEC is all-ones (grid exactly tiles
// the rows/cols; all dims are multiples of 16).
//
// VGPR layouts (wave32):
//   A 16x4 : lane l<16 -> A[m=l][k+0],A[m=l][k+1] ; l>=16 -> A[m=l-16][k+2..3]
//   B 4x16 : lane l<16 -> B[k+0][n=l],B[k+1][n=l] ; l>=16 -> B[k+2..3][n=l-16]
//   C/D    : VGPR r holds C[r + 8*(l>=16)][n = l&15]
// ---------------------------------------------------------------------------
__global__ void sage_gemm_wmma_kernel(const float* __restrict__ aggSum,
                                      const float* __restrict__ cnt,
                                      const float* __restrict__ root,
                                      const float* __restrict__ Wl,
                                      const float* __restrict__ Wr,
                                      const float* __restrict__ bias,
                                      float* __restrict__ out,
                                      int K, int ldOut, int applyRelu) {
    const int wave    = threadIdx.x >> 5;
    const int lane    = threadIdx.x & 31;
    const int l16     = lane & 15;
    const int half    = lane >> 4;               // 0: K lo pair, 1: K hi pair
    const int rowBase = blockIdx.x * 16;
    const int colTile = blockIdx.y * (blockDim.x >> 5) + wave;
    const int colBase = colTile * 16;

    const int m = rowBase + l16;                 // A row this lane loads
    const float scale = 1.0f / fmaxf(cnt[m], 1.0f);

    const float* aRow = aggSum + (size_t)m * K + 2 * half;
    const float* bRow = Wl + (size_t)(colBase + l16) * K + 2 * half;

    v8f c = {};
#pragma unroll 4
    for (int k = 0; k < K; k += 4) {
        v2f a, b;
        a.x = aRow[k] * scale;
        a.y = aRow[k + 1] * scale;
        b.x = bRow[k];
        b.y = bRow[k + 1];
        c = __builtin_amdgcn_wmma_f32_16x16x4_f32(false, a, false, b,
                                                  (short)0, c, false, false);
    }

    const float* a2 = root + (size_t)m * K + 2 * half;   // root features (ld==K)
    const float* b2 = Wr + (size_t)(colBase + l16) * K + 2 * half;
#pragma unroll 4
    for (int k = 0; k < K; k += 4) {
        v2f a, b;
        a.x = a2[k];
        a.y = a2[k + 1];
        b.x = b2[k];
        b.y = b2[k + 1];
        c = __builtin_amdgcn_wmma_f32_16x16x4_f32(false, a, false, b,
                                                  (short)0, c, false, false);
    }

    const float bv = bias[colBase + l16];
#pragma unroll
    for (int r = 0; r < 8; ++r) {
        float v = c[r] + bv;
        if (applyRelu) v = fmaxf(v, 0.0f);
        int row = rowBase + r + 8 * half;
        out[(size_t)row * ldOut + colBase + l16] = v;
    }
}

// ---------------------------------------------------------------------------
// In-place log_softmax over rows of 64. One wave per row (2 elems / lane).
// ---------------------------------------------------------------------------
__global__ void log_softmax64_kernel(float* __restrict__ h2, int rows) {
    int row = blockIdx.x * (blockDim.x >> 5) + (threadIdx.x >> 5);
    int lane = threadIdx.x & 31;
    if (row >= rows) return;
    float* p = h2 + (size_t)row * 64;
    float v0 = p[lane];
    float v1 = p[lane + 32];

    float m = fmaxf(v0, v1);
#pragma unroll
    for (int off = 16; off > 0; off >>= 1)
        m = fmaxf(m, __shfl_xor(m, off, 32));

    float s = __expf(v0 - m) + __expf(v1 - m);
#pragma unroll
    for (int off = 16; off > 0; off >>= 1)
        s += __shfl_xor(s, off, 32);

    float lse = m + __logf(s);
    p[lane]      = v0 - lse;
    p[lane + 32] = v1 - lse;
}

// ---------------------------------------------------------------------------
// Launch
// ---------------------------------------------------------------------------
extern "C" void kernel_launch(void* const* d_in, const int* in_sizes, int n_in,
                              void* d_out, int out_size, void* d_ws, size_t ws_size,
                              hipStream_t stream) {
    const float* x    = (const float*)d_in[0];
    const int*   src1 = (const int*)d_in[1];
    const int*   dst1 = (const int*)d_in[2];
    const int*   src2 = (const int*)d_in[3];
    const int*   dst2 = (const int*)d_in[4];
    const float* W_l1 = (const float*)d_in[5];
    const float* b_l1 = (const float*)d_in[6];
    const float* W_r1 = (const float*)d_in[7];
    const float* W_l2 = (const float*)d_in[8];
    const float* b_l2 = (const float*)d_in[9];
    const float* W_r2 = (const float*)d_in[10];
    float* out = (float*)d_out;

    // Workspace layout (floats): [agg1 | cnt1 | agg2 | cnt2 | h1]
    float* ws   = (float*)d_ws;
    float* agg1 = ws;                                  // N1*128
    float* cnt1 = agg1 + (size_t)N1_ * D_IN_;          // N1
    float* agg2 = cnt1 + N1_;                          // N2*256
    float* cnt2 = agg2 + (size_t)N2_ * D_H_;           // N2
    float* h1   = cnt2 + N2_;                          // N1*256
    const long zeroFloats = (long)N1_ * D_IN_ + N1_ + (long)N2_ * D_H_ + N2_;

    // 1) zero the accumulators (required every call: graph replays reuse ws)
    zero_f4_kernel<<<8192, 256, 0, stream>>>((float4*)ws, zeroFloats / 4);

    // 2) layer-1 scatter-sum: E1 edges, D_IN=128 (one wave per edge)
    scatter_sum_kernel<D_IN_><<<(E1_ + 7) / 8, 256, 0, stream>>>(
        x, src1, dst1, agg1, cnt1, E1_);

    // 3) h1 = relu(mean(agg1) @ W_l1^T + b_l1 + x[:N1] @ W_r1^T)  [N1 x 256]
    {
        dim3 grid(N1_ / 16, (D_H_ / 16) / 4);          // 6250 x 4 blocks
        sage_gemm_wmma_kernel<<<grid, 128, 0, stream>>>(
            agg1, cnt1, x, W_l1, W_r1, b_l1, h1, D_IN_, D_H_, 1);
    }

    // 4) layer-2 scatter-sum: E2 edges, D_H=256
    scatter_sum_kernel<D_H_><<<(E2_ + 7) / 8, 256, 0, stream>>>(
        h1, src2, dst2, agg2, cnt2, E2_);

    // 5) h2 = mean(agg2) @ W_l2^T + b_l2 + h1[:N2] @ W_r2^T  [N2 x 64] -> d_out
    {
        dim3 grid(N2_ / 16, 1);                        // 1250 x 1, 4 waves/block
        sage_gemm_wmma_kernel<<<grid, 128, 0, stream>>>(
            agg2, cnt2, h1, W_l2, W_r2, b_l2, out, D_H_, D_OUT_, 0);
    }

    // 6) in-place log_softmax over rows of 64
    log_softmax64_kernel<<<(N2_ + 7) / 8, 256, 0, stream>>>(out, N2_);
}